// TriMipEncoding_5669356830801
// MI455X (gfx1250) — compile-verified
//
#include <hip/hip_runtime.h>
#include <stdint.h>

// ---------------------------------------------------------------------------
// TriMipEncoding for MI455X (gfx1250).
// Memory-bound gather kernel: 192MB NT output stream + L2-resident texture
// (67MB < 192MB L2). Levels 5..7 (64.5KB) staged to LDS via the Tensor Data
// Mover (tensor_load_to_lds + s_wait_tensorcnt); LDS taps specialized so the
// compiler emits ds_load_b128 (not flat).
// ---------------------------------------------------------------------------

typedef float v4f __attribute__((ext_vector_type(4)));
typedef unsigned int v4u __attribute__((ext_vector_type(4)));
typedef int v4i __attribute__((ext_vector_type(4)));
typedef int v8i __attribute__((ext_vector_type(8)));

#define N_LEVELS 8
#define FEAT 16

// Float offsets of mip levels 1..7 inside the workspace (level 0 = input tex).
// sizes(floats): L1 3145728, L2 786432, L3 196608, L4 49152, L5 12288, L6 3072, L7 768
static constexpr int OFFS[8] = {0, 0, 3145728, 3932160, 4128768, 4177920, 4190208, 4193280};
#define LDS_FLOATS 16128           // levels 5..7 = 12288+3072+768 floats = 64512 bytes
#define LDS_QWORDS 8064            // 64512 / 8

#if __has_builtin(__builtin_amdgcn_tensor_load_to_lds) && __has_builtin(__builtin_amdgcn_s_wait_tensorcnt)
#define TRIMIP_USE_TDM 1
#endif

__device__ __forceinline__ int iclamp(int x, int lo, int hi) {
    return x < lo ? lo : (x > hi ? hi : x);
}

// ---------------------------------------------------------------------------
// Mip downsample: child level (res r = 1<<lr) from parent (res 2r), box
// filter. One thread per (plane, v, u, feat4); pure shift/mask indexing.
// ---------------------------------------------------------------------------
__global__ __launch_bounds__(256) void trimip_mip_down(const float* __restrict__ src,
                                                       float* __restrict__ dst, int lr) {
    int r = 1 << lr;
    int items = 3 << (2 * lr + 2);            // 3 * r * r * 4
    int idx = blockIdx.x * blockDim.x + threadIdx.x;
    if (idx >= items) return;
    int f4 = idx & 3;
    int t = idx >> 2;                         // (plane*r + v)*r + u
    int u = t & (r - 1);
    int v = (t >> lr) & (r - 1);
    int p = t >> (2 * lr);
    int R = r << 1;
    const float* b = src + ((size_t)(((p << lr | v) << 1) * R + 2 * u)) * FEAT + f4 * 4;
    // parent row index: (p*R + 2v) = ((p<<lr | v) << 1) since R = 2r.
    v4f a0 = *(const v4f*)b;
    v4f a1 = *(const v4f*)(b + FEAT);
    v4f a2 = *(const v4f*)(b + (size_t)R * FEAT);
    v4f a3 = *(const v4f*)(b + (size_t)R * FEAT + FEAT);
    v4f m = (a0 + a1 + a2 + a3) * 0.25f;
    *(v4f*)(dst + (size_t)t * FEAT + f4 * 4) = m;
}

// ---------------------------------------------------------------------------
// Bilinear tap: nvdiffrast texel-center convention with clamp, matches ref.
// Templated on pointer type so the LDS instantiation (pointer derived from a
// __shared__ array only) gets specialized to addrspace(3) -> ds_load_b128.
// ---------------------------------------------------------------------------
template <typename PtrT>
__device__ __forceinline__ v4f trimip_bilerp(PtrT base, int plane, int r,
                                             float u, float v, int f4) {
    float cu = u * (float)r - 0.5f;
    float cv = v * (float)r - 0.5f;
    float fu0 = floorf(cu), fv0 = floorf(cv);
    float fu = cu - fu0, fv = cv - fv0;
    int iu0 = iclamp((int)fu0, 0, r - 1);
    int iu1 = iclamp((int)fu0 + 1, 0, r - 1);
    int iv0 = iclamp((int)fv0, 0, r - 1);
    int iv1 = iclamp((int)fv0 + 1, 0, r - 1);
    int pr = plane * r;
    PtrT row0 = base + (size_t)(pr + iv0) * r * FEAT;
    PtrT row1 = base + (size_t)(pr + iv1) * r * FEAT;
    int c0 = iu0 * FEAT + f4 * 4;
    int c1 = iu1 * FEAT + f4 * 4;
    v4f t00 = *(const v4f*)(row0 + c0);
    v4f t01 = *(const v4f*)(row0 + c1);
    v4f t10 = *(const v4f*)(row1 + c0);
    v4f t11 = *(const v4f*)(row1 + c1);
    v4f top = t00 + (t01 - t00) * fu;
    v4f bot = t10 + (t11 - t10) * fu;
    return top + (bot - top) * fv;
}

// Global-memory base for levels 0..4.
__device__ __forceinline__ const float* trimip_gbase(int k, const float* tex0,
                                                     const float* mips) {
    return (k == 0) ? tex0 : (mips + OFFS[k]);
}

// ---------------------------------------------------------------------------
// Sampler: one thread per (point, plane, feat4). 4 consecutive lanes fetch one
// texel's 64B feature vector (coalesced gather); 12 lanes cover one point's
// 192B contiguous output (NT store, keeps texture resident in L2).
// ---------------------------------------------------------------------------
__global__ __launch_bounds__(256) void trimip_sample(const float* __restrict__ x,
                                                     const float* __restrict__ level,
                                                     const float* __restrict__ tex0,
                                                     const float* __restrict__ mips,
                                                     float* __restrict__ out,
                                                     int total) {
    __shared__ __align__(16) float s_mips[LDS_FLOATS];

#if defined(TRIMIP_USE_TDM)
    // Stage mips 5..7 (64512B, contiguous in ws at OFFS[5]) into LDS with one
    // TDM op, issued by wave 0 only (TDM ignores EXEC; issued once per wave).
    if (threadIdx.x < 32) {
        unsigned long long gaddr = (unsigned long long)(uintptr_t)(mips + OFFS[5]);
        unsigned ldsb = (unsigned)(uintptr_t)(void*)s_mips;  // flat low 32 bits == LDS offset

        v4u g0;
        g0.x = 1u;                                            // count=1 (valid descriptor)
        g0.y = ldsb;                                          // lds_addr [63:32]
        g0.z = (unsigned)(gaddr & 0xFFFFFFFFull);             // global_addr [95:64]
        g0.w = (unsigned)((gaddr >> 32) & 0x1FFFFFFull)       // global_addr [120:96]
               | (2u << 30);                                  // type=2 ("image")

        v8i g1;
        g1[0] = (3 << 16);                  // workgroup_mask=0, data_size=3 (8B)
        g1[1] = (LDS_QWORDS & 0xFFFF) << 16;// tensor_dim0 low16 @ bits 63:48
        g1[2] = (1 << 16);                  // tensor_dim0 hi16=0; tensor_dim1=1 low16
        g1[3] = (LDS_QWORDS & 0xFFFF) << 16;// tensor_dim1 hi16=0; tile_dim0 @ 127:112
        g1[4] = 1;                          // tile_dim1=1, tile_dim2=0
        g1[5] = LDS_QWORDS;                 // tensor_dim0_stride low32
        g1[6] = 0;                          // stride hi16=0; tensor_dim1_stride low16=0
        g1[7] = 0;

        v4i gz = {0, 0, 0, 0};              // groups 2/3 unused (<=2D tensor)
        v8i gz8 = {0, 0, 0, 0, 0, 0, 0, 0}; // extra descriptor payload (unused, zero)
        __builtin_amdgcn_tensor_load_to_lds(g0, g1, gz, gz, gz8, 0);
        __builtin_amdgcn_s_wait_tensorcnt(0);
    }
#else
    const float* src5 = mips + OFFS[5];
    for (int i = threadIdx.x; i < LDS_FLOATS; i += blockDim.x) s_mips[i] = src5[i];
#endif
    __syncthreads();

    const int stride = gridDim.x * blockDim.x;
    for (int idx = blockIdx.x * blockDim.x + threadIdx.x; idx < total; idx += stride) {
        int f4 = idx & 3;
        int t = idx >> 2;      // n*3 + plane
        int plane = t % 3;
        int n = t / 3;

        float lev = level[n];
        lev = fminf(fmaxf(lev, 0.0f), (float)(N_LEVELS - 1));
        float l0f = floorf(lev);
        int l0 = (int)l0f;
        float frac = lev - l0f;
        int l1 = l0 + 1 > (N_LEVELS - 1) ? (N_LEVELS - 1) : l0 + 1;

        float xx = x[3 * n + 0];
        float xy = x[3 * n + 1];
        float xz = x[3 * n + 2];
        float u = (plane == 0) ? xy : xx;
        float v = (plane == 2) ? xy : xz;

        v4f acc;
        if (l0 >= 5) {
            // Both taps come from the LDS-resident mips (l1 >= l0 >= 5).
            v4f a = trimip_bilerp(s_mips + (OFFS[l0] - OFFS[5]), plane, 512 >> l0, u, v, f4);
            v4f b = trimip_bilerp(s_mips + (OFFS[l1] - OFFS[5]), plane, 512 >> l1, u, v, f4);
            acc = a * (1.0f - frac) + b * frac;
        } else if (l1 >= 5) {
            // l0 == 4 (global), l1 == 5 (LDS).
            v4f a = trimip_bilerp(trimip_gbase(l0, tex0, mips), plane, 512 >> l0, u, v, f4);
            v4f b = trimip_bilerp(s_mips + 0, plane, 512 >> 5, u, v, f4);
            acc = a * (1.0f - frac) + b * frac;
        } else {
            // Both taps global (levels 0..4), L2-resident.
            v4f a = trimip_bilerp(trimip_gbase(l0, tex0, mips), plane, 512 >> l0, u, v, f4);
            v4f b = trimip_bilerp(trimip_gbase(l1, tex0, mips), plane, 512 >> l1, u, v, f4);
            acc = a * (1.0f - frac) + b * frac;
        }

        // out[n*48 + plane*16 + f] == out[t*16 + f4*4], streamed non-temporally.
        __builtin_nontemporal_store(acc, (v4f*)(out + (size_t)t * FEAT + f4 * 4));
    }
}

// ---------------------------------------------------------------------------
// Host launcher. Inputs: x (N*3 f32), level (N f32), texture (3*512*512*16 f32).
// d_ws holds the mip pyramid levels 1..7 (16,776,192 bytes needed).
// ---------------------------------------------------------------------------
extern "C" void kernel_launch(void* const* d_in, const int* in_sizes, int n_in,
                              void* d_out, int out_size, void* d_ws, size_t ws_size,
                              hipStream_t stream) {
    const float* x = (const float*)d_in[0];
    const float* level = (const float*)d_in[1];
    const float* tex = (const float*)d_in[2];
    float* out = (float*)d_out;
    float* mips = (float*)d_ws;

    int n = in_sizes[0] / 3;

    // Build mip cascade: level k from level k-1.
    const float* src = tex;
    for (int k = 1; k < N_LEVELS; ++k) {
        int lr = 9 - k;                      // log2(res of level k): 512>>k = 1<<(9-k)
        int r = 1 << lr;
        int items = 3 * r * r * 4;
        float* dst = mips + OFFS[k];
        trimip_mip_down<<<(items + 255) / 256, 256, 0, stream>>>(src, dst, lr);
        src = dst;
    }

    // Sampler: grid-stride so each workgroup amortizes its 64.5KB TDM staging
    // over ~3900 items. 3072 blocks * 256 threads (8 wave32s/WG).
    int total = n * 12;  // (point, plane, feat4)
    trimip_sample<<<3072, 256, 0, stream>>>(x, level, tex, mips, out, total);
}